// CRF_78804059947520
// MI455X (gfx1250) — compile-verified
//
#include <hip/hip_runtime.h>
#include <hip/hip_fp16.h>

// ---------------------------------------------------------------------------
// CRF forward (log-partition) for T=8192, K=1024, D=1024 on MI455X (gfx1250).
//
//  * gates: trivial GEMV kernel (34 MFLOP).
//  * scan : persistent 16-workgroup "cluster" kernel, 1 WG per WGP.
//           Each WG keeps its 64-row slice of the packed transition data
//           (f16 (ta-Cmax)*log2e | f16 (ti-ta)*log2e per element, 256 KB)
//           resident in CDNA5's 320 KB LDS for all 8192 steps.
//           Per step: fv -> F=max, vhat=exp2((fv-F)*log2e);
//           E = v_exp_f32(v_fma_mix_f32(g, dlt_f16, base_f16)) streamed
//           directly into the WMMA A-operand layout;
//           S[n] = sum_p E[n,p]*vhat[p] via V_WMMA_F32_16X16X4_F32;
//           fv' = feat + F + Cmax + ln2*v_log_f32(S).
//           Inter-WG sync: hardware cluster barrier when cluster-launched,
//           fenced agent-scope atomic barrier otherwise.
// ---------------------------------------------------------------------------

#define T_STEPS   8192
#define KDIM      1024
#define DDIM      1024
#define START_TAG (KDIM - 2)
#define STOP_TAG  (KDIM - 1)
#define NEGV      (-10000.0f)
#define L2E       1.4426950408889634f
#define LN2F      0.6931471805599453f
#define NWG       16

typedef float v2f __attribute__((ext_vector_type(2)));
typedef float v8f __attribute__((ext_vector_type(8)));

__device__ __forceinline__ unsigned ord_enc(float f) {
  unsigned u = __float_as_uint(f);
  return (u & 0x80000000u) ? ~u : (u | 0x80000000u);
}
__device__ __forceinline__ float ord_dec(unsigned u) {
  return (u & 0x80000000u) ? __uint_as_float(u & 0x7fffffffu) : __uint_as_float(~u);
}

// Raw hardware transcendentals: single v_exp_f32 / v_log_f32, no denorm fixup.
// Sub-2^-126 softmax contributions flushing to zero is exactly what we want.
__device__ __forceinline__ float exp2_raw(float x) { return __builtin_amdgcn_exp2f(x); }
__device__ __forceinline__ float log2_raw(float x) { return __builtin_amdgcn_logf(x); }

// One-instruction f16->f32 mixed FMA: d = fma(g, (f32)u.h[1], (f32)u.h[0])
//   S1 sel {op_sel_hi,op_sel} = {1,1} -> src[31:16] as f16 (delta)
//   S2 sel {op_sel_hi,op_sel} = {1,0} -> src[15:0]  as f16 (base)
__device__ __forceinline__ float fma_mix_hi_lo(float g, unsigned u) {
  float d;
  asm("v_fma_mix_f32 %0, %1, %2, %2 op_sel:[0,1,0] op_sel_hi:[0,1,1]"
      : "=v"(d)
      : "v"(g), "v"(u));
  return d;
}

// ---------------- init: reset reduction slot + grid-barrier counter --------
__global__ void k_init(unsigned* ordmax, unsigned* bar) {
  *ordmax = ord_enc(-__builtin_inff());
  *bar = 0u;
}

// ---------------- Cmax = max over max(ti, ta) ------------------------------
__global__ void k_maxtrans(const float* __restrict__ ti, const float* __restrict__ ta,
                           unsigned* ordmax) {
  __shared__ float red[256];
  int gid = blockIdx.x * 256 + threadIdx.x;
  float m = -__builtin_inff();
  for (int i = gid; i < KDIM * KDIM; i += 1024 * 256)
    m = fmaxf(m, fmaxf(ti[i], ta[i]));
  red[threadIdx.x] = m;
  __syncthreads();
  for (int s = 128; s > 0; s >>= 1) {
    if (threadIdx.x < s) red[threadIdx.x] = fmaxf(red[threadIdx.x], red[threadIdx.x + s]);
    __syncthreads();
  }
  if (threadIdx.x == 0) atomicMax(ordmax, ord_enc(red[0]));
}

// ---------------- pack transitions into WMMA-A-operand lane layout ---------
// Layout: P[wg(16)][tile(4)][chunk(256)][lane(32)][j(2)] (uint32 each)
//   lane = hi*16 + m ; element = row (wg*64 + tile*16 + m), col k = 4*chunk + 2*hi + j
//   dword = f16((ta-Cmax)*log2e)  |  f16((ti-ta)*log2e) << 16
__global__ void k_pack(const float* __restrict__ ti, const float* __restrict__ ta,
                       const unsigned* __restrict__ ordmax, unsigned* __restrict__ P) {
  int id = blockIdx.x * 256 + threadIdx.x;            // 0 .. 1,048,575
  float Cmax = ord_dec(*ordmax);
  int j    = id & 1;
  int lane = (id >> 1) & 31;
  int c    = (id >> 6) & 255;
  int tt   = (id >> 14) & 3;
  int wg   = id >> 16;
  int m = lane & 15, hi = lane >> 4;
  int n = wg * 64 + tt * 16 + m;
  int k = c * 4 + hi * 2 + j;
  size_t e = (size_t)n * KDIM + k;
  float base = (ta[e] - Cmax) * L2E;                  // masked entries -> -inf in f16 (E=0)
  float dlt  = (ti[e] - ta[e]) * L2E;                 // masks coincide, so dlt stays small
  unsigned ulo = (unsigned)__half_as_ushort(__float2half(base));
  unsigned uhi = (unsigned)__half_as_ushort(__float2half(dlt));
  P[id] = ulo | (uhi << 16);
}

// ---------------- gates: one wave per timestep -----------------------------
__global__ void k_gates(const float* __restrict__ reps, const float* __restrict__ wc,
                        const float* __restrict__ wu, const int* __restrict__ sc,
                        float* __restrict__ gate) {
  int wid = threadIdx.x >> 5, lane = threadIdx.x & 31;
  int t = blockIdx.x * 8 + wid;
  const float* cur = reps + (size_t)t * DDIM;
  const float* prv = reps + (size_t)(t > 0 ? t - 1 : 0) * DDIM;
  float ac = 0.f, au = 0.f;
  for (int d = lane; d < DDIM; d += 32) {
    float p = prv[d], q = cur[d];
    ac = fmaf(p, wc[d], ac); ac = fmaf(q, wc[DDIM + d], ac);
    au = fmaf(p, wu[d], au); au = fmaf(q, wu[DDIM + d], au);
  }
  for (int off = 16; off > 0; off >>= 1) {
    ac += __shfl_xor(ac, off, 32);
    au += __shfl_xor(au, off, 32);
  }
  if (lane == 0) {
    bool use_c = (sc[t] != 0) && (t > 0);
    float x = use_c ? ac : au;
    float g = 1.f / (1.f + __expf(-x));
    gate[t] = fminf(fmaxf(g, 1e-12f), 0.99999988f);
  }
}

// ---------------- grid-wide barrier (cluster HW barrier or atomic) ---------
__device__ __forceinline__ void grid_sync(unsigned* bar, unsigned* epoch) {
  __threadfence();                                    // release this WG's fv writes
  __syncthreads();
  if (threadIdx.x == 0) {
    if (__builtin_amdgcn_cluster_id_x() != 0) {
      __builtin_amdgcn_s_cluster_barrier();           // s_barrier_signal -3 ; s_barrier_wait -3
      __threadfence();                                // acquire
    } else {
      unsigned tgt = (unsigned)NWG * (++(*epoch));
      __hip_atomic_fetch_add(bar, 1u, __ATOMIC_RELEASE, __HIP_MEMORY_SCOPE_AGENT);
      while (__hip_atomic_load(bar, __ATOMIC_ACQUIRE, __HIP_MEMORY_SCOPE_AGENT) < tgt)
        __builtin_amdgcn_s_sleep(1);
    }
  }
  __syncthreads();
}

// ---------------- the sequential scan: 16 WGs, 256 threads each ------------
__global__ __launch_bounds__(256, 1)
void k_scan(const float* __restrict__ feats, const float* __restrict__ gate,
            const unsigned* __restrict__ P, const unsigned* __restrict__ ordmax,
            float* __restrict__ fvg, unsigned* bar) {
  extern __shared__ char smem_raw[];
  unsigned* Plds = (unsigned*)smem_raw;               // 65536 dwords (256 KB), persistent
  float* vhat = (float*)(smem_raw + 262144);          // 1024 floats
  float* Slds = vhat + KDIM;                          // 64 floats
  float* wmax = Slds + 64;                            // 8 floats

  const int tid  = threadIdx.x;
  const int wid  = tid >> 5, lane = tid & 31;
  const int tt   = wid >> 1;                          // 16-row tile within this WG (0..3)
  const int ks   = wid & 1;                           // k-split half (0..1)
  const int hi   = lane >> 4;
  const int wg   = blockIdx.x;
  const float Cmax = ord_dec(*ordmax);

  // One-time: copy this WG's 256 KB packed slice into LDS (stays all 8192 steps)
  {
    const uint4* src = (const uint4*)(P + (size_t)wg * 65536);
    uint4* dst = (uint4*)Plds;
    for (int i = tid; i < 16384; i += 256) dst[i] = src[i];
  }
  // Initial fv
  if (tid < 64) {
    int n = wg * 64 + tid;
    fvg[n] = (n == START_TAG) ? 0.f : NEGV;
  }
  __syncthreads();

  unsigned epoch = 0;
  const unsigned* Abase = Plds + ((size_t)((tt * 256 + ks * 128) * 32 + lane)) * 2;
  const float* vb = vhat + ks * 512 + hi * 2;

  for (int t = 0; t < T_STEPS; ++t) {
    grid_sync(bar, &epoch);                           // fv of previous step now visible

    // Phase 1: global max F of fv; vhat = exp2((fv - F)*log2e); zero S
    float fr0 = fvg[tid], fr1 = fvg[tid + 256], fr2 = fvg[tid + 512], fr3 = fvg[tid + 768];
    float m = fmaxf(fmaxf(fr0, fr1), fmaxf(fr2, fr3));
    for (int off = 16; off > 0; off >>= 1) m = fmaxf(m, __shfl_xor(m, off, 32));
    if (lane == 0) wmax[wid] = m;
    if (tid < 64) Slds[tid] = 0.f;
    __syncthreads();
    float F = wmax[0];
#pragma unroll
    for (int j2 = 1; j2 < 8; ++j2) F = fmaxf(F, wmax[j2]);
    vhat[tid]       = exp2_raw((fr0 - F) * L2E);
    vhat[tid + 256] = exp2_raw((fr1 - F) * L2E);
    vhat[tid + 512] = exp2_raw((fr2 - F) * L2E);
    vhat[tid + 768] = exp2_raw((fr3 - F) * L2E);
    __syncthreads();

    // Phase 2: E = exp2(fma_mix(g, dlt, base)) streamed into WMMA A-operand.
    const float g = gate[t];
    v8f acc = {0.f, 0.f, 0.f, 0.f, 0.f, 0.f, 0.f, 0.f};
#pragma unroll 8
    for (int c = 0; c < 128; ++c) {
      unsigned u0 = Abase[c * 64 + 0];                // ds_load_b64, conflict-free
      unsigned u1 = Abase[c * 64 + 1];
      float e0 = exp2_raw(fma_mix_hi_lo(g, u0));      // v_fma_mix_f32 + v_exp_f32
      float e1 = exp2_raw(fma_mix_hi_lo(g, u1));
      v2f A; A[0] = e0; A[1] = e1;                    // A[16x4]: lane=hi*16+m, K=2*hi+j
      v2f B; B[0] = vb[c * 4 + 0]; B[1] = vb[c * 4 + 1];  // B[4x16]: vhat broadcast over N
      acc = __builtin_amdgcn_wmma_f32_16x16x4_f32(false, A, false, B, (short)0, acc,
                                                  false, false);
    }
    // D[m, n] identical over n; lanes 0/16 hold rows tt*16 + hi*8 + j in acc[j]
    if ((lane & 15) == 0) {
      int rbase = tt * 16 + hi * 8;
#pragma unroll
      for (int j2 = 0; j2 < 8; ++j2) atomicAdd(&Slds[rbase + j2], acc[j2]); // ds_add_f32
    }
    __syncthreads();

    // Phase 3: fv' = feat + F + Cmax + ln2*log2(S)
    if (tid < 64) {
      int n = wg * 64 + tid;
      float s = Slds[tid];
      fvg[n] = feats[(size_t)t * KDIM + n] + F + Cmax + log2_raw(s) * LN2F;
      if (t + 1 < T_STEPS)
        __builtin_prefetch(feats + (size_t)(t + 1) * KDIM + n, 0, 0);  // global_prefetch_b8
    }
  }
}

// ---------------- terminal logsumexp ---------------------------------------
__global__ void k_final(const float* __restrict__ fvg, const float* __restrict__ reps,
                        const float* __restrict__ wu, const float* __restrict__ ti,
                        const float* __restrict__ ta, float* __restrict__ out) {
  __shared__ float red[256];
  int tid = threadIdx.x;
  // terminal gate g_t = sigmoid(reps[-1] . (wu[:D] + wu[D:]))
  float s = 0.f;
  for (int d = tid; d < DDIM; d += 256) {
    float r = reps[(size_t)(T_STEPS - 1) * DDIM + d];
    s = fmaf(r, wu[d] + wu[DDIM + d], s);
  }
  red[tid] = s; __syncthreads();
  for (int o = 128; o > 0; o >>= 1) { if (tid < o) red[tid] += red[tid + o]; __syncthreads(); }
  float gt = 1.f / (1.f + __expf(-red[0]));
  __syncthreads();

  float tv[4]; float m = -__builtin_inff();
#pragma unroll
  for (int j = 0; j < 4; ++j) {
    int n = tid + j * 256;
    float tr = gt * ti[(size_t)STOP_TAG * KDIM + n] + (1.f - gt) * ta[(size_t)STOP_TAG * KDIM + n];
    tv[j] = fvg[n] + tr;
    m = fmaxf(m, tv[j]);
  }
  red[tid] = m; __syncthreads();
  for (int o = 128; o > 0; o >>= 1) { if (tid < o) red[tid] = fmaxf(red[tid], red[tid + o]); __syncthreads(); }
  float M = red[0]; __syncthreads();
  float se = 0.f;
#pragma unroll
  for (int j = 0; j < 4; ++j) se += __expf(tv[j] - M);
  red[tid] = se; __syncthreads();
  for (int o = 128; o > 0; o >>= 1) { if (tid < o) red[tid] += red[tid + o]; __syncthreads(); }
  if (tid == 0) out[0] = M + logf(red[0]);
}

// ---------------------------------------------------------------------------
extern "C" void kernel_launch(void* const* d_in, const int* in_sizes, int n_in,
                              void* d_out, int out_size, void* d_ws, size_t ws_size,
                              hipStream_t stream) {
  const float* feats = (const float*)d_in[0];
  const float* reps  = (const float*)d_in[1];
  const float* wc    = (const float*)d_in[2];
  const float* wu    = (const float*)d_in[3];
  const float* ti    = (const float*)d_in[4];
  const float* ta    = (const float*)d_in[5];
  const int*   sc    = (const int*)d_in[6];
  float* out = (float*)d_out;

  char* ws = (char*)d_ws;
  unsigned* ordmax = (unsigned*)(ws + 0);
  unsigned* bar    = (unsigned*)(ws + 64);
  float*    fvg    = (float*)(ws + 128);        // 4 KB
  float*    gate   = (float*)(ws + 8192);       // 32 KB
  unsigned* P      = (unsigned*)(ws + 65536);   // 4 MB packed transitions

  size_t smem = 262144 + 4096 + 256 + 32;       // 256KB slice + vhat + S + wave-max
  (void)hipFuncSetAttribute((const void*)k_scan,
                            hipFuncAttributeMaxDynamicSharedMemorySize, (int)smem);

  k_init    <<<1,    1,   0, stream>>>(ordmax, bar);
  k_maxtrans<<<1024, 256, 0, stream>>>(ti, ta, ordmax);
  k_pack    <<<4096, 256, 0, stream>>>(ti, ta, ordmax, P);
  k_gates   <<<1024, 256, 0, stream>>>(reps, wc, wu, sc, gate);
  k_scan    <<<NWG,  256, smem, stream>>>(feats, gate, P, ordmax, fvg, bar);
  k_final   <<<1,    256, 0, stream>>>(fvg, reps, wu, ti, ta, out);
}